// SwitchWhiten2d_25769804234
// MI455X (gfx1250) — compile-verified
//
#include <hip/hip_runtime.h>
#include <hip/hip_bf16.h>

typedef float v2f __attribute__((ext_vector_type(2)));
typedef float v8f __attribute__((ext_vector_type(8)));

#define WMMA_F32(a, b, c) \
    __builtin_amdgcn_wmma_f32_16x16x4_f32(false, (a), false, (b), (short)0, (c), false, false)

// ---------------------------------------------------------------------------
// Kernel 1: per-(n,g) raw moments.  b = n*16+g, X_b is 16x4096 (rows = channels,
// contiguous, stride 4096).  S_b = X Xᵀ via V_WMMA_F32_16X16X4_F32 where the
// symmetric A/B fragments share the same registers; s_b = row sums via VALU.
// 512 blocks x 256 threads (8 waves); each wave covers 512 of the 4096 cols.
// ---------------------------------------------------------------------------
__global__ __launch_bounds__(256) void k_moments(const float* __restrict__ x,
                                                 float* __restrict__ S,
                                                 float* __restrict__ svec) {
    __shared__ float S_lds[256];
    __shared__ float s_lds[16];
    const int b    = blockIdx.x;
    const int tid  = threadIdx.x;
    const int lane = tid & 31;
    const int wave = tid >> 5;

    S_lds[tid] = 0.0f;
    if (tid < 16) s_lds[tid] = 0.0f;
    __syncthreads();

    const int row  = lane & 15;          // A-fragment M row (== C-fragment N col)
    const int koff = (lane >> 4) << 1;   // lanes 16-31 hold K+2, K+3
    const float* base = x + (size_t)b * 65536 + (size_t)row * 4096 + koff;

    v8f acc0 = {};
    v8f acc1 = {};
    float rs = 0.0f;
    const int k0 = wave * 512;
    // two independent WMMA accumulation chains for ILP
    for (int k = k0; k < k0 + 512; k += 8) {
        v2f f0 = *(const v2f*)(base + k);
        v2f f1 = *(const v2f*)(base + k + 4);
        acc0 = WMMA_F32(f0, f0, acc0);   // same regs as A and B: X·Xᵀ
        acc1 = WMMA_F32(f1, f1, acc1);
        rs += f0.x + f0.y + f1.x + f1.y;
    }
    v8f acc = acc0 + acc1;

    const int ro = (lane >> 4) << 3;     // C layout: row = r + 8*(lane>=16)
#pragma unroll
    for (int r = 0; r < 8; ++r)
        atomicAdd(&S_lds[(ro + r) * 16 + row], acc[r]);
    atomicAdd(&s_lds[row], rs);
    __syncthreads();

    S[(size_t)b * 256 + tid] = S_lds[tid];
    if (tid < 16) svec[b * 16 + tid] = s_lds[tid];
}

// ---------------------------------------------------------------------------
// Kernel 2a: reduce raw moments over the batch dimension per group.
// 16 blocks x 256 threads.
// ---------------------------------------------------------------------------
__global__ __launch_bounds__(256) void k_groupreduce(const float* __restrict__ S,
                                                     const float* __restrict__ svec,
                                                     float* __restrict__ Sg,
                                                     float* __restrict__ sg) {
    const int g = blockIdx.x;
    const int tid = threadIdx.x;
    float acc = 0.0f;
    for (int n = 0; n < 32; ++n)
        acc += S[(size_t)(n * 16 + g) * 256 + tid];
    Sg[g * 256 + tid] = acc;
    if (tid < 16) {
        float a2 = 0.0f;
        for (int n = 0; n < 32; ++n)
            a2 += svec[(n * 16 + g) * 16 + tid];
        sg[g * 16 + tid] = a2;
    }
}

// ---------------------------------------------------------------------------
// 16x16x16 matmul helper: D = A(16x16) * B(16x16), both row-major in LDS,
// as 4 chained V_WMMA_F32_16X16X4_F32.
// ---------------------------------------------------------------------------
__device__ inline v8f mm16(const float* A, const float* B, int lane) {
    v8f acc = {};
    const int m  = lane & 15;
    const int h2 = (lane >> 4) << 1;
#pragma unroll
    for (int kk = 0; kk < 4; ++kk) {
        const int ka = 4 * kk + h2;
        v2f a;  a.x  = A[m * 16 + ka];       a.y  = A[m * 16 + ka + 1];
        v2f bb; bb.x = B[ka * 16 + m];       bb.y = B[(ka + 1) * 16 + m];
        acc = WMMA_F32(a, bb, acc);
    }
    return acc;
}

__device__ inline void storeC(float* D, v8f acc, int lane) {
    const int col = lane & 15;
    const int ro  = (lane >> 4) << 3;
#pragma unroll
    for (int r = 0; r < 8; ++r)
        D[(ro + r) * 16 + col] = acc[r];
}

// ---------------------------------------------------------------------------
// Kernel 2b: per-batch stat mixing + Newton-Schulz iteration, all in WMMA.
// 512 blocks x 32 threads (one wave per 16x16 problem).
// ---------------------------------------------------------------------------
__global__ __launch_bounds__(32) void k_newton(const float* __restrict__ S,
                                               const float* __restrict__ svec,
                                               const float* __restrict__ Sg,
                                               const float* __restrict__ sg,
                                               const float* __restrict__ swm,
                                               const float* __restrict__ swv,
                                               float* __restrict__ wm_out,
                                               float* __restrict__ mean_out) {
    __shared__ float covN[256], P[256], T1[256], T2[256];
    __shared__ float mb[16], mi[16];
    const int b = blockIdx.x;
    const int g = b & 15;
    const int lane = threadIdx.x;

    // softmax over the two switch weights
    float e0 = __expf(swm[0]), e1 = __expf(swm[1]);
    const float mw0 = e0 / (e0 + e1), mw1 = e1 / (e0 + e1);
    e0 = __expf(swv[0]); e1 = __expf(swv[1]);
    const float vw0 = e0 / (e0 + e1), vw1 = e1 / (e0 + e1);

    const float invNM = 1.0f / (32.0f * 4096.0f);
    const float invM  = 1.0f / 4096.0f;

    if (lane < 16) {
        const float mbn = sg[g * 16 + lane] * invNM;
        const float min_ = svec[b * 16 + lane] * invM;
        mb[lane] = mbn;
        mi[lane] = min_;
        mean_out[b * 16 + lane] = mw0 * mbn + mw1 * min_;
    }
    __syncthreads();

    // cov = vw0*cov_bn + vw1*cov_in + eps*I  (each lane owns 8 strided elems)
    float tr_part = 0.0f;
#pragma unroll
    for (int r = 0; r < 8; ++r) {
        const int idx = lane + r * 32;
        const int i = idx >> 4, j = idx & 15;
        const float cbn = Sg[g * 256 + idx] * invNM - mb[i] * mb[j];
        const float cin = S[(size_t)b * 256 + idx] * invM - mi[i] * mi[j];
        float cv = vw0 * cbn + vw1 * cin;
        if (i == j) cv += 1e-5f;
        covN[idx] = cv;
        if (i == j) tr_part += cv;
    }
    for (int off = 16; off > 0; off >>= 1)
        tr_part += __shfl_xor(tr_part, off);
    const float rtr = 1.0f / tr_part;

#pragma unroll
    for (int r = 0; r < 8; ++r) {
        const int idx = lane + r * 32;
        covN[idx] *= rtr;
        P[idx] = ((idx >> 4) == (idx & 15)) ? 1.0f : 0.0f;
    }
    __syncthreads();

    const int col = lane & 15;
    const int ro  = (lane >> 4) << 3;
    for (int it = 0; it < 5; ++it) {
        v8f t1 = mm16(P, P, lane);
        storeC(T1, t1, lane);
        __syncthreads();
        v8f t2 = mm16(T1, P, lane);
        storeC(T2, t2, lane);
        __syncthreads();
        v8f t3 = mm16(T2, covN, lane);
        float np[8];
#pragma unroll
        for (int r = 0; r < 8; ++r)
            np[r] = 1.5f * P[(ro + r) * 16 + col] - 0.5f * t3[r];
        __syncthreads();
#pragma unroll
        for (int r = 0; r < 8; ++r)
            P[(ro + r) * 16 + col] = np[r];
        __syncthreads();
    }

    const float srtr = sqrtf(rtr);
#pragma unroll
    for (int r = 0; r < 8; ++r) {
        const int idx = lane + r * 32;
        wm_out[(size_t)b * 256 + idx] = P[idx] * srtr;
    }
}

// ---------------------------------------------------------------------------
// Kernel 4: x_hat = wm @ (X - mean), then affine.  512 blocks x 256 threads;
// each wave handles 32 column tiles of 16.  wm A-fragments + per-fragment
// means + per-row weight/bias are hoisted into registers before the tile loop.
// ---------------------------------------------------------------------------
__global__ __launch_bounds__(256) void k_whiten(const float* __restrict__ x,
                                                const float* __restrict__ wm,
                                                const float* __restrict__ mean,
                                                const float* __restrict__ weight,
                                                const float* __restrict__ bias,
                                                float* __restrict__ out) {
    __shared__ float wm_s[256];
    __shared__ float mean_s[16], w_s[16], b_s[16];
    const int b = blockIdx.x;
    const int g = b & 15;
    const int tid = threadIdx.x;
    const int lane = tid & 31;
    const int wave = tid >> 5;

    wm_s[tid] = wm[(size_t)b * 256 + tid];
    if (tid < 16) {
        mean_s[tid] = mean[b * 16 + tid];
        w_s[tid] = weight[g * 16 + tid];
        b_s[tid] = bias[g * 16 + tid];
    }
    __syncthreads();

    const int m  = lane & 15;
    const int h2 = (lane >> 4) << 1;
    const int ro = (lane >> 4) << 3;

    v2f av[4];
    float m0[4], m1[4];
#pragma unroll
    for (int kk = 0; kk < 4; ++kk) {
        const int ka = 4 * kk + h2;
        av[kk].x = wm_s[m * 16 + ka];
        av[kk].y = wm_s[m * 16 + ka + 1];
        m0[kk] = mean_s[ka];
        m1[kk] = mean_s[ka + 1];
    }
    float wr[8], br[8];
#pragma unroll
    for (int r = 0; r < 8; ++r) {
        wr[r] = w_s[ro + r];
        br[r] = b_s[ro + r];
    }

    const float* xg = x + (size_t)b * 65536;
    float* og = out + (size_t)b * 65536;

    for (int t = wave * 32; t < wave * 32 + 32; ++t) {
        const int cb = t * 16 + m;
        v8f acc = {};
#pragma unroll
        for (int kk = 0; kk < 4; ++kk) {
            const int ka = 4 * kk + h2;
            v2f bb;
            bb.x = xg[(size_t)ka * 4096 + cb] - m0[kk];
            bb.y = xg[(size_t)(ka + 1) * 4096 + cb] - m1[kk];
            acc = WMMA_F32(av[kk], bb, acc);
        }
#pragma unroll
        for (int r = 0; r < 8; ++r)
            og[(size_t)(ro + r) * 4096 + cb] = acc[r] * wr[r] + br[r];
    }
}

// ---------------------------------------------------------------------------
extern "C" void kernel_launch(void* const* d_in, const int* in_sizes, int n_in,
                              void* d_out, int out_size, void* d_ws, size_t ws_size,
                              hipStream_t stream) {
    const float* x      = (const float*)d_in[0];
    const float* swm    = (const float*)d_in[1];
    const float* swv    = (const float*)d_in[2];
    const float* weight = (const float*)d_in[3];
    const float* bias   = (const float*)d_in[4];
    float* out = (float*)d_out;

    float* ws   = (float*)d_ws;
    float* S    = ws;                 // 512*256
    float* sv   = S + 512 * 256;      // 512*16
    float* Sg   = sv + 512 * 16;      // 16*256
    float* sg   = Sg + 16 * 256;      // 16*16
    float* wmb  = sg + 16 * 16;       // 512*256
    float* mnb  = wmb + 512 * 256;    // 512*16

    k_moments<<<512, 256, 0, stream>>>(x, S, sv);
    k_groupreduce<<<16, 256, 0, stream>>>(S, sv, Sg, sg);
    k_newton<<<512, 32, 0, stream>>>(S, sv, Sg, sg, swm, swv, wmb, mnb);
    k_whiten<<<512, 256, 0, stream>>>(x, wmb, mnb, weight, bias, out);
}